// SparKDecoder_75797582840359
// MI455X (gfx1250) — compile-verified
//
#include <hip/hip_runtime.h>
#include <hip/hip_bf16.h>

// Problem constants (from the reference)
#define N_PTS 50000
#define M_TGT 8192
#define CHN   768
#define NWAVES 8
#define TILE_M 16
#define BATCH_SCALE 512.0f   // cross-batch penalty: S^2 = 262144 >> 3*127^2 = 48387
#define SAME_BATCH_THRESH 100000.0f

typedef __attribute__((ext_vector_type(2))) float v2f;
typedef __attribute__((ext_vector_type(8))) float v8f;

__device__ __forceinline__ unsigned long long shflXorU64(unsigned long long v, int mask) {
    unsigned lo = (unsigned)(v & 0xffffffffull);
    unsigned hi = (unsigned)(v >> 32);
    lo = __shfl_xor(lo, mask, 32);
    hi = __shfl_xor(hi, mask, 32);
    return (((unsigned long long)hi) << 32) | (unsigned long long)lo;
}

// One block = one tile of 16 targets. 8 waves stream all 50000 coords,
// computing 16x16 d^2 tiles with V_WMMA_F32_16X16X4_F32 (K=4: scaled-batch,x,y,z),
// tracking u64-packed (dist_bits, index) mins, then gathering features rows.
__global__ __launch_bounds__(256) void spark_nn_gather_kernel(
    const float* __restrict__ features,
    const int*   __restrict__ coords,
    const int*   __restrict__ target_coords,
    float*       __restrict__ out) {

    __shared__ unsigned long long partials[NWAVES][TILE_M];
    __shared__ int nearestIdx[TILE_M];

    const int tid  = threadIdx.x;
    const int lane = tid & 31;
    const int wv   = tid >> 5;
    const int half = (lane >> 4) & 1;     // 0: lanes 0-15, 1: lanes 16-31
    const int l16  = lane & 15;
    const int t0   = blockIdx.x * TILE_M;

    const int4* tc4 = reinterpret_cast<const int4*>(target_coords); // (b,x,y,z)
    const int4* cc4 = reinterpret_cast<const int4*>(coords);

    // ---- A matrix: 16x4 targets, pre-scaled by -2 so WMMA accumulates -2*t.c ----
    // Layout (ISA 16x4 f32 A): lanes 0-15 hold K=0,1 in v0,v1; lanes 16-31 hold K=2,3.
    int4 tA = tc4[t0 + l16];
    float tv0 = BATCH_SCALE * (float)tA.x;
    float tv1 = (float)tA.y;
    float tv2 = (float)tA.z;
    float tv3 = (float)tA.w;
    v2f a;
    a.x = -2.0f * (half ? tv2 : tv0);
    a.y = -2.0f * (half ? tv3 : tv1);

    // ---- per-lane target norms matching the C/D layout (VGPR v -> M = v + 8*half) ----
    float tn[8];
    #pragma unroll
    for (int v = 0; v < 8; ++v) {
        int4 t = tc4[t0 + v + half * 8];
        float b0 = BATCH_SCALE * (float)t.x;
        float x = (float)t.y, y = (float)t.z, z = (float)t.w;
        tn[v] = b0 * b0 + x * x + y * y + z * z;
    }

    unsigned long long best[8];
    #pragma unroll
    for (int v = 0; v < 8; ++v) best[v] = ~0ull;

    // ---- stream coord chunks: wave wv handles chunks wv, wv+8, ... (N % 16 == 0) ----
    for (int j0 = wv * 16; j0 < N_PTS; j0 += 16 * NWAVES) {
        int4 cc = cc4[j0 + l16];                        // column n = l16 of this chunk
        float cb = BATCH_SCALE * (float)cc.x;
        float cx = (float)cc.y, cy = (float)cc.z, cz = (float)cc.w;
        float cn = cb * cb + cx * cx + cy * cy + cz * cz;

        // B matrix 4x16: lanes 0-15 carry rows K=0,1; lanes 16-31 rows K=2,3 (selects, no branch)
        v2f b;
        b.x = half ? cy : cb;
        b.y = half ? cz : cx;

        // Preload accumulator with ||t||^2 + ||c||^2; WMMA adds -2*t.c -> exact d^2
        v8f c;
        #pragma unroll
        for (int v = 0; v < 8; ++v) c[v] = tn[v] + cn;

        c = __builtin_amdgcn_wmma_f32_16x16x4_f32(
                /*neg_a=*/false, a, /*neg_b=*/false, b,
                /*c_mod=*/(short)0, c, /*reuse_a=*/false, /*reuse_b=*/false);

        unsigned j = (unsigned)(j0 + l16);
        #pragma unroll
        for (int v = 0; v < 8; ++v) {
            // d^2 >= 0 and exact (all integers < 2^24) -> f32 bits are order-preserving
            unsigned long long key =
                (((unsigned long long)__float_as_uint(c[v])) << 32) | (unsigned long long)j;
            best[v] = (key < best[v]) ? key : best[v];
        }
    }

    // ---- cross-lane min inside each 16-lane half (halves hold disjoint M rows) ----
    #pragma unroll
    for (int mask = 8; mask >= 1; mask >>= 1) {
        #pragma unroll
        for (int v = 0; v < 8; ++v) {
            unsigned long long o = shflXorU64(best[v], mask);
            best[v] = (o < best[v]) ? o : best[v];
        }
    }
    if (l16 == 0) {
        #pragma unroll
        for (int v = 0; v < 8; ++v) partials[wv][half * 8 + v] = best[v];
    }
    __syncthreads();

    // ---- cross-wave min + decode ----
    if (tid < TILE_M) {
        unsigned long long k = partials[0][tid];
        #pragma unroll
        for (int w = 1; w < NWAVES; ++w) {
            unsigned long long o = partials[w][tid];
            k = (o < k) ? o : k;
        }
        float d2 = __uint_as_float((unsigned)(k >> 32));
        int idx  = (int)(unsigned)(k & 0xffffffffull);
        nearestIdx[tid] = (d2 < SAME_BATCH_THRESH) ? idx : -1;   // -1 => empty batch -> zeros
    }
    __syncthreads();

    // ---- gather 16 rows x 768 f32 with b128 (float4) loads/stores ----
    const float4* f4 = reinterpret_cast<const float4*>(features);
    float4*       o4 = reinterpret_cast<float4*>(out);
    const int C4 = CHN / 4;                        // 192
    for (int e = tid; e < TILE_M * C4; e += 256) {
        int row = e / C4;
        int col = e - row * C4;
        int idx = nearestIdx[row];
        float4 val;
        if (idx >= 0) {
            val = f4[(long long)idx * C4 + col];
        } else {
            val.x = 0.0f; val.y = 0.0f; val.z = 0.0f; val.w = 0.0f;
        }
        o4[(long long)(t0 + row) * C4 + col] = val;
    }
}

extern "C" void kernel_launch(void* const* d_in, const int* in_sizes, int n_in,
                              void* d_out, int out_size, void* d_ws, size_t ws_size,
                              hipStream_t stream) {
    (void)in_sizes; (void)n_in; (void)out_size; (void)d_ws; (void)ws_size;
    const float* features      = (const float*)d_in[0];   // N x C f32
    const int*   coords        = (const int*)d_in[1];     // N x 4 i32
    const int*   target_coords = (const int*)d_in[2];     // M x 4 i32
    float*       out           = (float*)d_out;           // M x C f32

    dim3 grid(M_TGT / TILE_M);   // 512 blocks
    dim3 block(256);             // 8 wave32
    spark_nn_gather_kernel<<<grid, block, 0, stream>>>(features, coords, target_coords, out);
}